// TopKRouter_59064390254753
// MI455X (gfx1250) — compile-verified
//
#include <hip/hip_runtime.h>
#include <hip/hip_bf16.h>

typedef __attribute__((ext_vector_type(2))) float v2f;
typedef __attribute__((ext_vector_type(4))) float v4f;
typedef __attribute__((ext_vector_type(8))) float v8f;

#define BB 4
#define SS 4096
#define DD 2048
#define KTOP 819            // int(4096 * 0.2)
#define NROWS (BB * SS)     // 16384 logits

#define TARGET_SPARSITY 0.2f
#define SPARSITY_COEF 0.1f
#define ENTROPY_COEF 0.01f
#define EPS_ 1e-10f

// ---------------------------------------------------------------------------
// Kernel 1: logits[r] = dot(x[r, :], gate_w) + gate_b  via fp32 WMMA.
// One wave per 16-row tile. Per 8-K chunk each lane loads one b128 of x and
// one b128 of LDS-staged w; two WMMAs consume (x0,x1|x4,x5) and (x2,x3|x6,x7)
// under a consistent A/B K-permutation (valid: WMMA sums over K). Four
// rotating accumulators break the D->C dependency chain (128 deep each).
// ---------------------------------------------------------------------------
__global__ __launch_bounds__(256) void
gemv_wmma_f32(const float* __restrict__ x, const float* __restrict__ w,
              const float* __restrict__ bias, float* __restrict__ logits) {
    __shared__ float ws[DD];                    // 8 KB staged gate_w
    const int tid = threadIdx.x;

    // Coalesced b128 staging of gate_w into LDS (2048 floats / 256 thr / 4).
#pragma unroll
    for (int j = 0; j < DD / (256 * 4); ++j) {
        int i4 = tid + j * 256;
        ((v4f*)ws)[i4] = ((const v4f*)w)[i4];
    }
    __syncthreads();

    const int lane = tid & 31;
    const int wave = tid >> 5;
    const int tile = blockIdx.x * 8 + wave;     // 1024 tiles of 16 rows
    const int row_base = tile * 16;
    const int m    = lane & 15;                 // A-matrix row for this lane
    const int half = lane >> 4;                 // 0: K slots {0,1}, 1: {2,3}

    const v4f* __restrict__ xr4 = (const v4f*)(x + (size_t)(row_base + m) * DD);
    const v4f* __restrict__ w4  = (const v4f*)ws;

    v8f acc0 = {}, acc1 = {}, acc2 = {}, acc3 = {};
    for (int k8 = 0; k8 < DD / 8; k8 += 4) {    // 4 chunks of 8 K per trip
#pragma unroll
        for (int c = 0; c < 4; ++c) {
            const int base4 = (k8 + c) * 2 + half;  // half0: k..k+3, half1: k+4..k+7
            v4f xv = xr4[base4];
            v4f wv = w4[base4];
            v2f a0 = {xv.x, xv.y}, b0 = {wv.x, wv.y};
            v2f a1 = {xv.z, xv.w}, b1 = {wv.z, wv.w};
            v8f& acc = (c == 0) ? acc0 : (c == 1) ? acc1 : (c == 2) ? acc2 : acc3;
            acc = __builtin_amdgcn_wmma_f32_16x16x4_f32(
                false, a0, false, b0, (short)0, acc, false, false);
            acc = __builtin_amdgcn_wmma_f32_16x16x4_f32(
                false, a1, false, b1, (short)0, acc, false, false);
        }
    }
    v8f accf = (acc0 + acc1) + (acc2 + acc3);

    // D layout: VGPR j -> M=j (lanes 0-15), M=j+8 (lanes 16-31); all columns
    // equal, so lane 0 writes rows 0..7 and lane 16 writes rows 8..15.
    if (m == 0) {
        const float b0 = bias[0];
        const int r = row_base + half * 8;
#pragma unroll
        for (int j = 0; j < 8; ++j)
            logits[r + j] = accf[j] + b0;
    }
}

// ---------------------------------------------------------------------------
// Kernel 2: aux_loss = 0.1*(mean(p)-0.2)^2 + 0.01*mean(binary_entropy(p))
// ---------------------------------------------------------------------------
__global__ __launch_bounds__(1024) void
aux_loss_kernel(const float* __restrict__ logits, float* __restrict__ aux_out) {
    __shared__ float sp[1024];
    __shared__ float se[1024];
    const int tid = threadIdx.x;

    float sum_p = 0.0f, sum_e = 0.0f;
    for (int i = tid; i < NROWS; i += 1024) {
        float l = logits[i];
        float p = 1.0f / (1.0f + __expf(-l));
        sum_p += p;
        sum_e += -(p * __logf(p + EPS_) + (1.0f - p) * __logf(1.0f - p + EPS_));
    }
    sp[tid] = sum_p;
    se[tid] = sum_e;
    __syncthreads();
    for (int off = 512; off > 0; off >>= 1) {
        if (tid < off) {
            sp[tid] += sp[tid + off];
            se[tid] += se[tid + off];
        }
        __syncthreads();
    }
    if (tid == 0) {
        const float inv_n = 1.0f / (float)NROWS;
        float mean_p = sp[0] * inv_n;
        float dsl = mean_p - TARGET_SPARSITY;
        float sparsity_loss = dsl * dsl;
        float ent_mean = se[0] * inv_n;
        aux_out[0] = SPARSITY_COEF * sparsity_loss + ENTROPY_COEF * ent_mean;
    }
}

// ---------------------------------------------------------------------------
// Kernel 3: per-row top-819 select -> mask (0/1 floats) + sorted indices.
// ---------------------------------------------------------------------------
__device__ __forceinline__ int block_excl_scan(int v, int* scan_buf, int tid) {
    scan_buf[tid] = v;
    __syncthreads();
    for (int off = 1; off < 1024; off <<= 1) {
        int mine = scan_buf[tid];
        int add  = (tid >= off) ? scan_buf[tid - off] : 0;
        __syncthreads();
        scan_buf[tid] = mine + add;
        __syncthreads();
    }
    return scan_buf[tid] - v;   // exclusive prefix
}

__global__ __launch_bounds__(1024) void
topk_kernel(const float* __restrict__ logits, const float* __restrict__ log_temp,
            float* __restrict__ mask_out, float* __restrict__ idx_out) {
    __shared__ unsigned int keys[SS];
    __shared__ int hist[256];
    __shared__ int scan_buf[1024];
    __shared__ int sh_rem, sh_digit;

    const int b   = blockIdx.x;
    const int tid = threadIdx.x;

    float temp = __expf(log_temp[0]);
    temp = fminf(fmaxf(temp, 0.1f), 2.0f);
    const float inv_t = 1.0f / temp;

    const float* __restrict__ lr = logits + b * SS;

    // Load noisy logits as sign-flipped orderable u32 keys.
#pragma unroll
    for (int j = 0; j < SS / 1024; ++j) {
        int s = tid + j * 1024;
        float v = lr[s] * inv_t;
        unsigned int u = __float_as_uint(v);
        keys[s] = (u & 0x80000000u) ? ~u : (u | 0x80000000u);
    }
    if (tid == 0) sh_rem = KTOP;
    __syncthreads();

    // Radix-select k-th largest key, 8 bits per pass, MSB first.
    unsigned int prefix = 0, pmask = 0;
    for (int shift = 24; shift >= 0; shift -= 8) {
        for (int i = tid; i < 256; i += 1024) hist[i] = 0;
        __syncthreads();
#pragma unroll
        for (int i = 0; i < 4; ++i) {
            unsigned int key = keys[tid * 4 + i];
            if ((key & pmask) == prefix)
                atomicAdd(&hist[(key >> shift) & 255u], 1);
        }
        __syncthreads();
        if (tid == 0) {
            int rem = sh_rem;
            int d = 255;
            while (d > 0 && hist[d] < rem) { rem -= hist[d]; --d; }
            sh_digit = d;
            sh_rem = rem;
        }
        __syncthreads();
        prefix |= ((unsigned int)sh_digit) << shift;
        pmask  |= (0xFFu << shift);
        __syncthreads();
    }
    const unsigned int T = prefix;       // k-th largest key value
    const int needed_eq  = sh_rem;       // how many ==T to take (lowest index)

    unsigned int k4[4];
    int cnt_eq = 0;
#pragma unroll
    for (int i = 0; i < 4; ++i) {
        k4[i] = keys[tid * 4 + i];
        cnt_eq += (k4[i] == T);
    }

    // Scan (a): exclusive prefix of ==T counts -> tie-break rank.
    int eq_rank = block_excl_scan(cnt_eq, scan_buf, tid);

    int sel[4];
    int cnt_sel = 0;
#pragma unroll
    for (int i = 0; i < 4; ++i) {
        bool gt = (k4[i] > T);
        bool eq = (k4[i] == T);
        bool take = gt || (eq && (eq_rank < needed_eq));
        if (eq) ++eq_rank;
        sel[i] = take ? 1 : 0;
        cnt_sel += sel[i];
    }
    __syncthreads();

    // Scan (b): exclusive prefix of selected -> ascending output slot.
    int pos = block_excl_scan(cnt_sel, scan_buf, tid);

    float* __restrict__ mrow = mask_out + b * SS;
    float* __restrict__ irow = idx_out + b * KTOP;
#pragma unroll
    for (int i = 0; i < 4; ++i) {
        int s = tid * 4 + i;
        mrow[s] = sel[i] ? 1.0f : 0.0f;
        if (sel[i]) {
            irow[pos] = (float)s;   // contiguous-ownership scan => sorted
            ++pos;
        }
    }
}

// ---------------------------------------------------------------------------
// Launch: outputs concatenated [mask(16384) | indices(3276) | logits(16384) | aux(1)]
// ---------------------------------------------------------------------------
extern "C" void kernel_launch(void* const* d_in, const int* in_sizes, int n_in,
                              void* d_out, int out_size, void* d_ws, size_t ws_size,
                              hipStream_t stream) {
    const float* x        = (const float*)d_in[0];
    const float* gate_w   = (const float*)d_in[1];
    const float* gate_b   = (const float*)d_in[2];
    const float* log_temp = (const float*)d_in[3];

    float* out   = (float*)d_out;
    float* mask  = out;                             // 16384
    float* idx   = out + NROWS;                     // 3276
    float* logit = out + NROWS + BB * KTOP;         // 16384
    float* aux   = out + NROWS + BB * KTOP + NROWS; // 1

    // 1024 tiles of 16 rows, 8 waves/block -> 128 blocks.
    gemv_wmma_f32<<<NROWS / 16 / 8, 256, 0, stream>>>(x, gate_w, gate_b, logit);
    aux_loss_kernel<<<1, 1024, 0, stream>>>(logit, aux);
    topk_kernel<<<BB, 1024, 0, stream>>>(logit, log_temp, mask, idx);
}